// BoundaryLoss_77283641524662
// MI455X (gfx1250) — compile-verified
//
#include <hip/hip_runtime.h>
#include <hip/hip_bf16.h>

#define BB 4
#define HH 512
#define WW 512
#define HW (HH * WW)
#define NPART (BB * HH)

typedef __attribute__((ext_vector_type(2))) float v2f;
typedef __attribute__((ext_vector_type(8))) float v8f;

#if defined(__HIP_DEVICE_COMPILE__)
#if __has_builtin(__builtin_amdgcn_global_load_async_to_lds_b32)
#define USE_ASYNC_LDS 1
#endif
#if __has_builtin(__builtin_amdgcn_wmma_f32_16x16x4_f32)
#define USE_WMMA_REDUCE 1
#endif
#endif

// ---------------------------------------------------------------------------
// Zero the per-image foreground counters (ws is not re-poisoned per replay).
// ---------------------------------------------------------------------------
__global__ __launch_bounds__(64) void k_init(int* __restrict__ fgCnt) {
    if (threadIdx.x < BB) fgCnt[threadIdx.x] = 0;
}

// ---------------------------------------------------------------------------
// Pass 1: per-(image, column) 1-D nearest-seed distance (forward + backward
// scan), for both seed sets: bg seeds (target != 1) -> pos_dist,
// fg seeds (target == 1) -> neg_dist. g stored as u16, 0xFFFF = "no seed".
// Also counts fg pixels per image (integer atomics: deterministic).
// ---------------------------------------------------------------------------
__global__ __launch_bounds__(256) void k_cols(const int* __restrict__ tgt,
                                              unsigned short* __restrict__ gbuf,
                                              int* __restrict__ fgCnt) {
    int gid = blockIdx.x * blockDim.x + threadIdx.x;
    if (gid >= BB * WW) return;
    int b = gid / WW, j = gid % WW;
    const int* t = tgt + (size_t)b * HW;
    unsigned short* gB = gbuf + (size_t)(b * 2 + 0) * HW;  // bg seeds -> pos
    unsigned short* gF = gbuf + (size_t)(b * 2 + 1) * HW;  // fg seeds -> neg
    int lastBg = -1, lastFg = -1, cnt = 0;
    for (int i = 0; i < HH; ++i) {
        int fg = (t[i * WW + j] == 1);
        cnt += fg;
        if (fg) lastFg = i; else lastBg = i;
        gF[i * WW + j] = (unsigned short)(lastFg >= 0 ? (i - lastFg) : 0xFFFF);
        gB[i * WW + j] = (unsigned short)(lastBg >= 0 ? (i - lastBg) : 0xFFFF);
    }
    int nxtBg = -1, nxtFg = -1;
    for (int i = HH - 1; i >= 0; --i) {
        int fg = (t[i * WW + j] == 1);
        if (fg) nxtFg = i; else nxtBg = i;
        unsigned dF = gF[i * WW + j];
        unsigned dN = (nxtFg >= 0) ? (unsigned)(nxtFg - i) : 0xFFFFu;
        gF[i * WW + j] = (unsigned short)(dF < dN ? dF : dN);
        dF = gB[i * WW + j];
        dN = (nxtBg >= 0) ? (unsigned)(nxtBg - i) : 0xFFFFu;
        gB[i * WW + j] = (unsigned short)(dF < dN ? dF : dN);
    }
    atomicAdd(&fgCnt[b], cnt);
}

// ---------------------------------------------------------------------------
// Pass 2: one block per (image, row). Stage both g-rows into LDS (async
// global->LDS on CDNA5), convert once to interleaved f32 g^2, then the
// min-plus lower envelope over k (VALU-bound inner loop fed by ds_load_b64
// broadcast), fused with softmax weighting and block reduction.
// ---------------------------------------------------------------------------
__global__ __launch_bounds__(256) void k_rows(const float* __restrict__ pred,
                                              const unsigned short* __restrict__ gbuf,
                                              const int* __restrict__ fgCnt,
                                              float* __restrict__ partials) {
    __shared__ float g2s[WW][2];   // [k][0]=bg seed g^2, [k][1]=fg seed g^2
    __shared__ float red[256];
#if USE_ASYNC_LDS
    __shared__ unsigned int stage0[WW / 2];
    __shared__ unsigned int stage1[WW / 2];
#endif
    const int tid = threadIdx.x;
    const int blk = blockIdx.x;
    const int b = blk / HH, i = blk % HH;

    const unsigned int* row0 =
        (const unsigned int*)(gbuf + (size_t)(b * 2 + 0) * HW + (size_t)i * WW);
    const unsigned int* row1 =
        (const unsigned int*)(gbuf + (size_t)(b * 2 + 1) * HW + (size_t)i * WW);

    unsigned v0, v1;
#if USE_ASYNC_LDS
    {
        // builtin signature (from hipcc diagnostic): param0 is AS(1) int*
        typedef __attribute__((address_space(1))) int* gp_t;
        typedef __attribute__((address_space(3))) int* lp_t;
        __builtin_amdgcn_global_load_async_to_lds_b32((gp_t)(row0 + tid),
                                                      (lp_t)&stage0[tid], 0, 0);
        __builtin_amdgcn_global_load_async_to_lds_b32((gp_t)(row1 + tid),
                                                      (lp_t)&stage1[tid], 0, 0);
#if __has_builtin(__builtin_amdgcn_s_wait_asynccnt)
        __builtin_amdgcn_s_wait_asynccnt(0);
#else
        asm volatile("s_wait_asynccnt 0" ::: "memory");
#endif
        __syncthreads();
        v0 = stage0[tid];
        v1 = stage1[tid];
    }
#else
    v0 = row0[tid];
    v1 = row1[tid];
#endif
    // unpack 2 u16 per u32; g<=511 so g*g exact in f32; 0xFFFF -> INF^2=1e18
    unsigned a0 = v0 & 0xFFFFu, a1 = v0 >> 16;
    unsigned c0 = v1 & 0xFFFFu, c1 = v1 >> 16;
    g2s[2 * tid + 0][0] = (a0 == 0xFFFFu) ? 1e18f : (float)(a0 * a0);
    g2s[2 * tid + 1][0] = (a1 == 0xFFFFu) ? 1e18f : (float)(a1 * a1);
    g2s[2 * tid + 0][1] = (c0 == 0xFFFFu) ? 1e18f : (float)(c0 * c0);
    g2s[2 * tid + 1][1] = (c1 == 0xFFFFu) ? 1e18f : (float)(c1 * c1);
    __syncthreads();

    // each thread owns columns j0 = tid and j1 = tid+256; shared k-loop so one
    // ds_load_b64 feeds 4 fma+4 min (VALU-bound by design).
    const int j0 = tid, j1 = tid + 256;
    float d0 = (float)j0, d1 = (float)j1;
    float pB0 = 1e30f, pF0 = 1e30f, pB1 = 1e30f, pF1 = 1e30f;
#pragma unroll 4
    for (int k = 0; k < WW; ++k) {
        float gB = g2s[k][0];
        float gF = g2s[k][1];
        pB0 = fminf(pB0, fmaf(d0, d0, gB));
        pF0 = fminf(pF0, fmaf(d0, d0, gF));
        pB1 = fminf(pB1, fmaf(d1, d1, gB));
        pF1 = fminf(pF1, fmaf(d1, d1, gF));
        d0 -= 1.0f;
        d1 -= 1.0f;
    }
    const float present = (fgCnt[b] > 0) ? 1.0f : 0.0f;
    float dm0 = (sqrtf(pF0) - sqrtf(pB0)) * present;  // neg - pos
    float dm1 = (sqrtf(pF1) - sqrtf(pB1)) * present;

    // softmax over 2 classes: p1 = sigmoid(x1 - x0); class-0 dmap is 0.
    const float* pr = pred + ((size_t)(b * 2) * HH + i) * WW;  // class 0 row
    float x00 = pr[j0], x01 = pr[j1];
    float x10 = pr[HW + j0], x11 = pr[HW + j1];                // class 1 row
    float s0 = 1.0f / (1.0f + __expf(x00 - x10));
    float s1 = 1.0f / (1.0f + __expf(x01 - x11));

    red[tid] = s0 * dm0 + s1 * dm1;
    __syncthreads();
    for (int off = 128; off > 0; off >>= 1) {
        if (tid < off) red[tid] += red[tid + off];
        __syncthreads();
    }
    if (tid == 0) partials[blk] = red[0];
}

// ---------------------------------------------------------------------------
// Final reduce: one wave. Fixed-order serial sums per lane, then a WMMA
// ones-matmul (D[M,N] = sum_K A[M,K]) folds the 32 lane partials exactly.
// ---------------------------------------------------------------------------
__global__ __launch_bounds__(32) void k_final(const float* __restrict__ partials,
                                              float* __restrict__ out) {
    const int lane = threadIdx.x;
    float s = 0.0f;
    for (int t = lane; t < NPART; t += 32) s += partials[t];
    float tot;
#if USE_WMMA_REDUCE
    // A (16x4 f32): vgpr0 lanes0-15 = A[M,0]=s_M, lanes16-31 = A[M,2]=s_{M+16};
    // vgpr1 = 0. B = ones => D[M,N] = s_M + s_{M+16} for all N.
    v2f a;  a[0] = s;    a[1] = 0.0f;
    v2f bn; bn[0] = 1.0f; bn[1] = 1.0f;
    v8f c = {0.0f, 0.0f, 0.0f, 0.0f, 0.0f, 0.0f, 0.0f, 0.0f};
    v8f d = __builtin_amdgcn_wmma_f32_16x16x4_f32(false, a, false, bn, (short)0,
                                                  c, false, false);
    // lane 0 holds D[M=0..7, N=0] in its 8 vgprs, lane 16 holds D[M=8..15, N=0]
    float r = ((d[0] + d[1]) + (d[2] + d[3])) + ((d[4] + d[5]) + (d[6] + d[7]));
    tot = __shfl(r, 0, 32) + __shfl(r, 16, 32);
#else
    float r = s;
    for (int off = 16; off > 0; off >>= 1) r += __shfl_xor(r, off, 32);
    tot = r;
#endif
    if (lane == 0) out[0] = tot * (1.0f / (float)((size_t)BB * 2 * HH * WW));
}

extern "C" void kernel_launch(void* const* d_in, const int* in_sizes, int n_in,
                              void* d_out, int out_size, void* d_ws, size_t ws_size,
                              hipStream_t stream) {
    const float* pred = (const float*)d_in[0];       // [B, 2, H, W] f32
    const int* target = (const int*)d_in[1];         // [B, H, W] i32

    // workspace layout: g (u16, 4MB) | fgCnt (4 ints) | partials (2048 f32)
    unsigned short* gbuf = (unsigned short*)d_ws;
    size_t goff = (size_t)BB * 2 * HW * sizeof(unsigned short);
    int* fgCnt = (int*)((char*)d_ws + goff);
    float* partials = (float*)((char*)d_ws + goff + 64);

    k_init<<<1, 64, 0, stream>>>(fgCnt);
    k_cols<<<(BB * WW + 255) / 256, 256, 0, stream>>>(target, gbuf, fgCnt);
    k_rows<<<BB * HH, 256, 0, stream>>>(pred, gbuf, fgCnt, partials);
    k_final<<<1, 32, 0, stream>>>(partials, (float*)d_out);
}